// GCN_47588237639689
// MI455X (gfx1250) — compile-verified
//
#include <hip/hip_runtime.h>

typedef float v2f __attribute__((ext_vector_type(2)));
typedef float v8f __attribute__((ext_vector_type(8)));

#define N_NODES 100000
#define N_EDGES 600000
#define DIM     128
#define NPART   391          // ceil(100000/256)

// ============ CSR construction ============

__global__ void csr_clear(int* __restrict__ p, int n) {
    int i = blockIdx.x * blockDim.x + threadIdx.x;
    if (i < n) p[i] = 0;
}

__global__ void csr_degree(const long long* __restrict__ dst,
                           int* __restrict__ degI, int nE) {
    int e = blockIdx.x * blockDim.x + threadIdx.x;
    if (e < nE) atomicAdd(&degI[(int)dst[e]], 1);
}

// per-block exclusive scan; block total -> partial[b]
__global__ __launch_bounds__(256)
void csr_scan_block(const int* __restrict__ degI,
                    int* __restrict__ rowstart,
                    int* __restrict__ partial) {
    __shared__ int sh[256];
    const int tid = threadIdx.x;
    const int i   = blockIdx.x * 256 + tid;
    const int v   = (i < N_NODES) ? degI[i] : 0;
    sh[tid] = v;
    __syncthreads();
#pragma unroll
    for (int off = 1; off < 256; off <<= 1) {
        int t = (tid >= off) ? sh[tid - off] : 0;
        __syncthreads();
        sh[tid] += t;
        __syncthreads();
    }
    if (i < N_NODES) rowstart[i] = sh[tid] - v;     // local exclusive
    if (tid == 255)  partial[blockIdx.x] = sh[tid]; // block total
}

// single block: exclusive scan of NPART partials (padded to 512)
__global__ __launch_bounds__(512)
void csr_scan_partials(int* __restrict__ partial) {
    __shared__ int sh[512];
    const int tid = threadIdx.x;
    const int v   = (tid < NPART) ? partial[tid] : 0;
    sh[tid] = v;
    __syncthreads();
#pragma unroll
    for (int off = 1; off < 512; off <<= 1) {
        int t = (tid >= off) ? sh[tid - off] : 0;
        __syncthreads();
        sh[tid] += t;
        __syncthreads();
    }
    if (tid < NPART) partial[tid] = sh[tid] - v;    // exclusive
}

__global__ void csr_scan_add(int* __restrict__ rowstart,
                             const int* __restrict__ partial) {
    int i = blockIdx.x * blockDim.x + threadIdx.x;
    if (i < N_NODES) rowstart[i] += partial[i >> 8];
    if (i == 0) rowstart[N_NODES] = N_EDGES;
}

__global__ void csr_cursor_copy(const int* __restrict__ rowstart,
                                int* __restrict__ cursor, int n) {
    int i = blockIdx.x * blockDim.x + threadIdx.x;
    if (i < n) cursor[i] = rowstart[i];
}

__global__ void csr_fill(const long long* __restrict__ src,
                         const long long* __restrict__ dst,
                         int* __restrict__ cursor,
                         int* __restrict__ col, int nE) {
    int e = blockIdx.x * blockDim.x + threadIdx.x;
    if (e < nE) {
        int d   = (int)dst[e];
        int pos = atomicAdd(&cursor[d], 1);
        col[pos] = (int)src[e];
    }
}

// ============ gather-reduce mean aggregation (no atomics) ============
// one wave32 per node; lane holds 4 of the 128 features
__global__ __launch_bounds__(256)
void sage_gather(const int* __restrict__ rowstart,
                 const int* __restrict__ col,
                 const float* __restrict__ h,
                 float* __restrict__ agg) {     // [N,128], pre-scaled by 1/deg
    const int wave = (int)((blockIdx.x * blockDim.x + threadIdx.x) >> 5);
    const int lane = threadIdx.x & 31;
    if (wave >= N_NODES) return;                 // exact grid: never taken
    const int start = rowstart[wave];
    const int end   = rowstart[wave + 1];
    const float inv = 1.0f / fmaxf((float)(end - start), 1.0f);

    float ax = 0.f, ay = 0.f, az = 0.f, aw = 0.f;
    for (int base = start; base < end; base += 32) {
        int cnt = end - base;
        if (cnt > 32) cnt = 32;
        const int ci = (lane < cnt) ? col[base + lane] : 0;
        for (int j = 0; j < cnt; ++j) {
            const int s = __shfl(ci, j, 32);
            const float4 v = *(const float4*)(h + (size_t)s * DIM + lane * 4);
            ax += v.x; ay += v.y; az += v.z; aw += v.w;
        }
    }
    float4 o; o.x = ax * inv; o.y = ay * inv; o.z = az * inv; o.w = aw * inv;
    *(float4*)(agg + (size_t)wave * DIM + lane * 4) = o;
}

// ============ fused SAGE layer: out = relu(agg @ Wl + bl + h @ Wr) ============
// block = 256 thr = 8 waves; block owns 16 rows (LDS-staged), wave w owns cols [16w,16w+16)
#define LPAD 132   // 128 + 4 pad: lane m reads bank (4m + ka) -> conflict-free

__global__ __launch_bounds__(256)
void sage_gemm(const float* __restrict__ agg,   // [N,128] pre-scaled mean
               const float* __restrict__ h,     // [N,128]
               const float* __restrict__ Wl,    // [128,128] row-major [k][n]
               const float* __restrict__ bl,    // [128]
               const float* __restrict__ Wr,    // [128,128]
               float* __restrict__ out) {       // [N,128]
    __shared__ float sA[16][LPAD];
    __shared__ float sH[16][LPAD];

    const int tid   = threadIdx.x;
    const int mbase = blockIdx.x * 16;

    // cooperative stage of the 16 A-rows of both matrices (2 x 8 KB)
    {
        const int r  = tid >> 4;           // 0..15
        const int c  = (tid & 15) * 8;     // 0..120 step 8
        const float* pa = agg + (size_t)(mbase + r) * DIM + c;
        const float* ph = h   + (size_t)(mbase + r) * DIM + c;
        *(float4*)&sA[r][c]     = *(const float4*)pa;
        *(float4*)&sA[r][c + 4] = *(const float4*)(pa + 4);
        *(float4*)&sH[r][c]     = *(const float4*)ph;
        *(float4*)&sH[r][c + 4] = *(const float4*)(ph + 4);
    }
    __syncthreads();

    const int lane  = tid & 31;
    const int wv    = tid >> 5;            // 0..7 -> N tile
    const int nbase = wv * 16;
    const int mrow  = lane & 15;           // A: M row; B/C: N col
    const int hi    = lane >> 4;           // K half select

    v8f c = {};
    for (int k = 0; k < DIM; k += 4) {
        const int ka = k + 2 * hi;         // per-lane K base (A/B layout)
        const v2f a0 = *(const v2f*)&sA[mrow][ka];
        v2f b0; b0.x = Wl[(size_t)ka * DIM + nbase + mrow];
                b0.y = Wl[(size_t)(ka + 1) * DIM + nbase + mrow];
        c = __builtin_amdgcn_wmma_f32_16x16x4_f32(false, a0, false, b0,
                                                  (short)0, c, false, false);
        const v2f a1 = *(const v2f*)&sH[mrow][ka];
        v2f b1; b1.x = Wr[(size_t)ka * DIM + nbase + mrow];
                b1.y = Wr[(size_t)(ka + 1) * DIM + nbase + mrow];
        c = __builtin_amdgcn_wmma_f32_16x16x4_f32(false, a1, false, b1,
                                                  (short)0, c, false, false);
    }

    const float bias = bl[nbase + mrow];   // bias depends only on N col
    // C layout: VGPR r -> M = r + 8*hi, N = lane&15
#pragma unroll
    for (int r = 0; r < 8; ++r) {
        const int row = mbase + r + 8 * hi;
        out[(size_t)row * DIM + nbase + mrow] = fmaxf(c[r] + bias, 0.0f);
    }
}

// ============ lin head: relu(relu(h @ W0 + b0) @ W1 + b1) -> [N,8] ============
__global__ __launch_bounds__(128)
void lin_head(const float* __restrict__ h,    // [N,128]
              const float* __restrict__ W0,   // [128,64]
              const float* __restrict__ b0,   // [64]
              const float* __restrict__ W1,   // [64,8]
              const float* __restrict__ b1,   // [8]
              float* __restrict__ out) {      // [N,8]
    __shared__ float t[16][64 + 4];

    const int lane  = threadIdx.x & 31;
    const int wv    = threadIdx.x >> 5;    // 0..3 -> N tile of lin0
    const int mbase = blockIdx.x * 16;
    const int mrow  = lane & 15;
    const int hi    = lane >> 4;
    const int nbase = wv * 16;

    const float* hRow = h + (size_t)(mbase + mrow) * DIM;

    v8f c = {};
    for (int k = 0; k < DIM; k += 4) {
        const int ka = k + 2 * hi;
        v2f a; a.x = hRow[ka]; a.y = hRow[ka + 1];
        v2f b; b.x = W0[(size_t)ka * 64 + nbase + mrow];
               b.y = W0[(size_t)(ka + 1) * 64 + nbase + mrow];
        c = __builtin_amdgcn_wmma_f32_16x16x4_f32(false, a, false, b,
                                                  (short)0, c, false, false);
    }
    const float bias = b0[nbase + mrow];
#pragma unroll
    for (int r = 0; r < 8; ++r)
        t[r + 8 * hi][nbase + mrow] = fmaxf(c[r] + bias, 0.0f);

    __syncthreads();

    if (wv == 0) {                          // whole wave enters: EXEC all-ones
        const float nmask = (mrow < 8) ? 1.0f : 0.0f;   // branchless col mask
        const int   ncl   = (mrow < 8) ? mrow : 0;
        v8f c2 = {};
        for (int k = 0; k < 64; k += 4) {
            const int ka = k + 2 * hi;
            v2f a; a.x = t[mrow][ka]; a.y = t[mrow][ka + 1];
            v2f b; b.x = W1[(size_t)ka * 8 + ncl] * nmask;
                   b.y = W1[(size_t)(ka + 1) * 8 + ncl] * nmask;
            c2 = __builtin_amdgcn_wmma_f32_16x16x4_f32(false, a, false, b,
                                                       (short)0, c2, false, false);
        }
        const float bias2 = b1[ncl];
        if (mrow < 8) {
#pragma unroll
            for (int r = 0; r < 8; ++r) {
                const int row = mbase + r + 8 * hi;
                out[(size_t)row * 8 + mrow] = fmaxf(c2[r] + bias2, 0.0f);
            }
        }
    }
}

// ============ launch ============

extern "C" void kernel_launch(void* const* d_in, const int* in_sizes, int n_in,
                              void* d_out, int out_size, void* d_ws, size_t ws_size,
                              hipStream_t stream) {
    const float*     x    = (const float*)d_in[0];
    const long long* eidx = (const long long*)d_in[1];   // [2, 600000] int64
    const float* s0_Wl = (const float*)d_in[2];
    const float* s0_bl = (const float*)d_in[3];
    const float* s0_Wr = (const float*)d_in[4];
    const float* s1_Wl = (const float*)d_in[5];
    const float* s1_bl = (const float*)d_in[6];
    const float* s1_Wr = (const float*)d_in[7];
    const float* l0_W  = (const float*)d_in[8];
    const float* l0_b  = (const float*)d_in[9];
    const float* l1_W  = (const float*)d_in[10];
    const float* l1_b  = (const float*)d_in[11];
    float* out = (float*)d_out;

    const long long* src = eidx;             // edge_index[0]
    const long long* dst = eidx + N_EDGES;   // edge_index[1]

    // workspace carve-up (256B aligned)
    char* ws = (char*)d_ws;
    auto carve = [&](size_t bytes) {
        char* p = ws;
        ws += (bytes + 255) & ~(size_t)255;
        return p;
    };
    int*   degI     = (int*)carve(sizeof(int) * N_NODES);
    int*   rowstart = (int*)carve(sizeof(int) * (N_NODES + 1));
    int*   partial  = (int*)carve(sizeof(int) * 512);
    int*   cursor   = (int*)carve(sizeof(int) * N_NODES);
    int*   col      = (int*)carve(sizeof(int) * N_EDGES);
    float* agg      = (float*)carve(sizeof(float) * (size_t)N_NODES * DIM);
    float* h1       = (float*)carve(sizeof(float) * (size_t)N_NODES * DIM);
    float* h2       = (float*)carve(sizeof(float) * (size_t)N_NODES * DIM);

    dim3 b256(256), b128(128), b512(512);
    dim3 gNode((N_NODES + 255) / 256);       // 391
    dim3 gEdge((N_EDGES + 255) / 256);
    dim3 gWaveNode(N_NODES / 8);             // 12500 (exact: 1 wave/node)
    dim3 gTile(N_NODES / 16);                // 6250  (exact)

    // ---- CSR build (edge list reused by both layers) ----
    csr_clear        <<<gNode, b256, 0, stream>>>(degI, N_NODES);
    csr_degree       <<<gEdge, b256, 0, stream>>>(dst, degI, N_EDGES);
    csr_scan_block   <<<NPART, b256, 0, stream>>>(degI, rowstart, partial);
    csr_scan_partials<<<1,     b512, 0, stream>>>(partial);
    csr_scan_add     <<<gNode, b256, 0, stream>>>(rowstart, partial);
    csr_cursor_copy  <<<gNode, b256, 0, stream>>>(rowstart, cursor, N_NODES);
    csr_fill         <<<gEdge, b256, 0, stream>>>(src, dst, cursor, col, N_EDGES);

    // ---- layer 0 ----
    sage_gather<<<gWaveNode, b256, 0, stream>>>(rowstart, col, x, agg);
    sage_gemm  <<<gTile,     b256, 0, stream>>>(agg, x, s0_Wl, s0_bl, s0_Wr, h1);

    // ---- layer 1 ----
    sage_gather<<<gWaveNode, b256, 0, stream>>>(rowstart, col, h1, agg);
    sage_gemm  <<<gTile,     b256, 0, stream>>>(agg, h1, s1_Wl, s1_bl, s1_Wr, h2);

    // ---- MLP head ----
    lin_head   <<<gTile,     b128, 0, stream>>>(h2, l0_W, l0_b, l1_W, l1_b, out);

    (void)in_sizes; (void)n_in; (void)out_size; (void)ws_size;
}